// SurfaceConv_3822520893769
// MI455X (gfx1250) — compile-verified
//
#include <hip/hip_runtime.h>

#define B_   8
#define N_   4096
#define P_   4096
#define KNB  32
#define C_   128
#define WOUT 16
#define EPSV 1e-5f

typedef __attribute__((ext_vector_type(2))) float v2f;
typedef __attribute__((ext_vector_type(8))) float v8f;

// ---------------------------------------------------------------------------
// Fused: gather + point-weights + stage-3 einsum (WMMA) + stage-5 GEMM (WMMA)
// grid: 256 blocks x 128 threads (4 waves); each wave owns 32 points of batch b.
// Produces pre-norm y[b][p][c] (B,P,128) in workspace.
// ---------------------------------------------------------------------------
__global__ void __launch_bounds__(128)
fused_surfaceconv(const float* __restrict__ points,
                  const float* __restrict__ lc,
                  const int*   __restrict__ nbr,
                  const int*   __restrict__ didx,
                  const float* __restrict__ Ww,
                  const float* __restrict__ bw,
                  const float* __restrict__ Wl,
                  const float* __restrict__ bl,
                  float*       __restrict__ y)
{
    extern __shared__ float smem[];
    const int tid  = threadIdx.x;
    const int wave = tid >> 5;
    const int lane = tid & 31;
    const int w    = lane & 15;
    const int hi   = lane >> 4;           // 0: lanes 0-15, 1: lanes 16-31
    float* F = smem + wave * (32 * 256);  // 32 KB per wave: [32 pts][256 K-chunk]

    const int b     = blockIdx.x >> 5;                 // 32 blocks per batch
    const int pbase = (blockIdx.x & 31) * 128 + wave * 32;
    const int sb    = didx[b];                         // batch selection (data_idx)

    // per-lane column of Ww and bias (w = output weight-channel of this lane)
    const float ww0 = Ww[0 * WOUT + w];
    const float ww1 = Ww[1 * WOUT + w];
    const float ww2 = Ww[2 * WOUT + w];
    const float bwv = bw[w];

    v8f acc[16];                                       // [mt*8+nt], 128 VGPRs
    #pragma unroll
    for (int i = 0; i < 16; ++i) acc[i] = v8f{};

    for (int cb = 0; cb < 8; ++cb) {                   // 8 channel blocks of 16
        const int c = cb * 16 + w;

        // ---- phase 1: stage-3 einsum  M_p(16c x 16w) = D_p^T @ W_p  -------
        for (int pl = 0; pl < 32; ++pl) {
            const int p       = pbase + pl;
            const int rowbase = (sb * P_ + p) * KNB;
            v8f fa = v8f{};
            #pragma unroll
            for (int kk = 0; kk < 8; ++kk) {           // K=32 in steps of 4
                const int kA = kk * 4 + hi * 2;        // A layout: lo lanes K=0,1 ; hi lanes K=2,3
                const int n0 = nbr[rowbase + kA];
                const int n1 = nbr[rowbase + kA + 1];
                v2f a, bb;
                a.x = points[(sb * N_ + n0) * C_ + c]; // gathered A fragment
                a.y = points[(sb * N_ + n1) * C_ + c];
                const float* lcr = lc + (size_t)(rowbase + kA) * 3;
                bb.x = fmaf(lcr[0], ww0, fmaf(lcr[1], ww1, fmaf(lcr[2], ww2, bwv)));
                bb.y = fmaf(lcr[3], ww0, fmaf(lcr[4], ww1, fmaf(lcr[5], ww2, bwv)));
                fa = __builtin_amdgcn_wmma_f32_16x16x4_f32(
                         false, a, false, bb, (short)0, fa, false, false);
            }
            // scatter D-layout result into F[pl][c_local*16 + w]
            float* Fp = F + pl * 256 + hi * (8 * 16) + w;
            #pragma unroll
            for (int r = 0; r < 8; ++r) Fp[r * 16] = fa[r];
        }
        __syncthreads();

        // ---- phase 2: stage-5 GEMM over this K-chunk (256 of 2048) --------
        for (int ks = 0; ks < 64; ++ks) {
            const int k0 = ks * 4 + hi * 2;            // local K
            const int kg = cb * 256 + k0;              // global Wl row
            v2f aA, aB;
            aA.x = F[w * 256 + k0];                    // point rows 0-15
            aA.y = F[w * 256 + k0 + 1];
            aB.x = F[(16 + w) * 256 + k0];             // point rows 16-31
            aB.y = F[(16 + w) * 256 + k0 + 1];
            const float* wl0 = Wl + (size_t)kg * C_;
            #pragma unroll
            for (int nt = 0; nt < 8; ++nt) {
                const int nn = nt * 16 + w;
                v2f bb;
                bb.x = wl0[nn];
                bb.y = wl0[C_ + nn];
                acc[nt]     = __builtin_amdgcn_wmma_f32_16x16x4_f32(
                                  false, aA, false, bb, (short)0, acc[nt],     false, false);
                acc[8 + nt] = __builtin_amdgcn_wmma_f32_16x16x4_f32(
                                  false, aB, false, bb, (short)0, acc[8 + nt], false, false);
            }
        }
        __syncthreads();
    }

    // ---- epilogue: add bias bl, store pre-norm y[b][p][j] -----------------
    #pragma unroll
    for (int mt = 0; mt < 2; ++mt) {
        #pragma unroll
        for (int nt = 0; nt < 8; ++nt) {
            const int   j   = nt * 16 + w;
            const float blj = bl[j];
            #pragma unroll
            for (int r = 0; r < 8; ++r) {
                const int p = pbase + mt * 16 + r + hi * 8;   // D layout row
                y[(size_t)(b * P_ + p) * C_ + j] = acc[mt * 8 + nt][r] + blj;
            }
        }
    }
}

// ---------------------------------------------------------------------------
// Per-channel mean / rsqrt(var+eps) over all B*P samples.
// 128 blocks (one per channel) x 256 threads.
// ---------------------------------------------------------------------------
__global__ void __launch_bounds__(256)
bn_stats(const float* __restrict__ y, float* __restrict__ mean, float* __restrict__ rinv)
{
    const int c   = blockIdx.x;
    const int tid = threadIdx.x;
    float s = 0.f, s2 = 0.f;
    for (int i = tid; i < B_ * P_; i += 256) {
        float v = y[(size_t)i * C_ + c];
        s  += v;
        s2 += v * v;
    }
    __shared__ float sh[512];
    sh[tid] = s; sh[256 + tid] = s2;
    __syncthreads();
    for (int off = 128; off > 0; off >>= 1) {
        if (tid < off) { sh[tid] += sh[tid + off]; sh[256 + tid] += sh[256 + tid + off]; }
        __syncthreads();
    }
    if (tid == 0) {
        const float inv_n = 1.0f / (float)(B_ * P_);
        float m   = sh[0] * inv_n;
        float var = sh[256] * inv_n - m * m;
        mean[c] = m;
        rinv[c] = rsqrtf(var + EPSV);
    }
}

// ---------------------------------------------------------------------------
// Normalize + gamma/beta + ReLU, writing transposed (B,C,P) layout.
// ---------------------------------------------------------------------------
__global__ void __launch_bounds__(256)
bn_apply(const float* __restrict__ y, const float* __restrict__ mean,
         const float* __restrict__ rinv, const float* __restrict__ gamma,
         const float* __restrict__ beta, float* __restrict__ outx)
{
    const int idx = blockIdx.x * 256 + threadIdx.x;   // over B*C*P
    const int p = idx & (P_ - 1);
    const int c = (idx >> 12) & (C_ - 1);
    const int b = idx >> 19;
    const float v = y[(size_t)(b * P_ + p) * C_ + c];
    const float r = (v - mean[c]) * rinv[c] * gamma[c] + beta[c];
    outx[idx] = r > 0.f ? r : 0.f;
}

__global__ void __launch_bounds__(256)
copy_xyz(const float* __restrict__ xyz, float* __restrict__ out)
{
    const int idx = blockIdx.x * 256 + threadIdx.x;
    out[idx] = xyz[idx];
}

// ---------------------------------------------------------------------------
extern "C" void kernel_launch(void* const* d_in, const int* in_sizes, int n_in,
                              void* d_out, int out_size, void* d_ws, size_t ws_size,
                              hipStream_t stream)
{
    const float* xyz    = (const float*)d_in[0];
    const float* points = (const float*)d_in[1];
    const float* lc     = (const float*)d_in[2];
    const int*   nbr    = (const int*)  d_in[3];
    const int*   didx   = (const int*)  d_in[4];
    const float* Ww     = (const float*)d_in[5];
    const float* bw     = (const float*)d_in[6];
    const float* Wl     = (const float*)d_in[7];
    const float* bl     = (const float*)d_in[8];
    const float* gamma  = (const float*)d_in[9];
    const float* beta   = (const float*)d_in[10];

    float* out  = (float*)d_out;
    float* y    = (float*)d_ws;                      // (B,P,128) pre-norm
    float* mean = y + (size_t)B_ * P_ * C_;
    float* rinv = mean + C_;

    const int lds_bytes = 4 * 32 * 256 * (int)sizeof(float);   // 128 KB dynamic
    (void)hipFuncSetAttribute((const void*)fused_surfaceconv,
                              hipFuncAttributeMaxDynamicSharedMemorySize, lds_bytes);

    fused_surfaceconv<<<256, 128, lds_bytes, stream>>>(points, lc, nbr, didx,
                                                       Ww, bw, Wl, bl, y);
    bn_stats<<<C_, 256, 0, stream>>>(y, mean, rinv);
    bn_apply<<<(B_ * C_ * P_) / 256, 256, 0, stream>>>(y, mean, rinv, gamma, beta,
                                                       out + (size_t)B_ * P_ * 3);
    copy_xyz<<<(B_ * P_ * 3) / 256, 256, 0, stream>>>(xyz, out);
}